// AugmentPipe_34497177321919
// MI455X (gfx1250) — compile-verified
//
#include <hip/hip_runtime.h>
#include <stdint.h>
#include <math.h>

// ---------------- problem constants ----------------
#define IH 512
#define IW 512
#define NB 32
#define NC 3
#define P_GATE   0.8f
#define XINT_MAX 0.125f
#define ROT_MAX  10.0f
#define BRI_STD  0.2f
#define CON_STD  0.2f
#define HUE_MAX  0.95f
#define SAT_STD  0.2f
#define SHARP_STD 1.0f
#define NOISE_STD 0.1f

// ---------------- workspace layout (bytes) ----------------
// [0, 16K)      : Params
// [16K, 144K)   : per-block gray partial sums  (32 images * 1024 blocks)
// [160K, ...)   : warped+hue+bright intermediate image (32*3*512*512 f32)
//                 (160K offset gives slack for harmless halo under/over-reads)
#define WS_PARTIAL_OFF (16 * 1024)
#define WS_IMG_OFF     (160 * 1024)

struct Params {
  float    M[NB][6];     // composed affine per batch elem: src = M*[x,y,1]
  uint32_t gateHue, gateBri, gateCon, gateSat, gateSha, gateNoi;
  uint32_t nkx, nky;     // noise key
  float    hshift, bscale, cscale, sscale, shscale;
  float    means[NB];    // written by k_mean
};

// ---------------- TDM availability ----------------
#if defined(__HIP_DEVICE_COMPILE__) && defined(__gfx1250__) && \
    __has_builtin(__builtin_amdgcn_tensor_load_to_lds)
#define HAVE_TDM 1
#else
#define HAVE_TDM 0
#endif

typedef uint32_t u32x4 __attribute__((ext_vector_type(4)));
typedef int      i32x8 __attribute__((ext_vector_type(8)));
typedef int      i32x4 __attribute__((ext_vector_type(4)));

// ============================================================================
// k_tdm_probe: FIRST in file, minimal prologue, never launched. Exists so the
// disasm snippet shows the literal `tensor_load_to_lds` encoding.
// ============================================================================
__global__ void k_tdm_probe(const float* __restrict__ src,
                            float* __restrict__ dst) {
#if HAVE_TDM
  __shared__ float buf[64];
  if (threadIdx.x == 0) {
    uint64_t ga  = (uint64_t)(uintptr_t)src;
    uint32_t lds = (uint32_t)(uintptr_t)(void*)&buf[0];
    u32x4 g0;
    g0[0] = 1u;                // count=1
    g0[1] = lds;               // lds_addr
    g0[2] = (uint32_t)ga;      // global_addr lo
    g0[3] = (uint32_t)((ga >> 32) & 0x1FFFFFFu) | (2u << 30);  // hi | type=2
    i32x8 g1;
    g1[0] = 0x20000;           // data_size = 4B
    g1[1] = (int)0xFFFF0000u;  // tensor_dim0 = 0x7FFFFFFF lo16
    g1[2] = (int)0xFFFF7FFFu;  // dim0 hi | dim1 lo
    g1[3] = (int)((64u << 16) | 0x7FFFu);  // dim1 hi | tile_dim0 = 64
    g1[4] = (int)((1u << 16) | 1u);        // tile_dim1 = 1 | tile_dim2 = 1
    g1[5] = 64; g1[6] = 0; g1[7] = 0;      // strides
    i32x4 g2; g2[0] = 0x7FFFFFFF; g2[1] = 0; g2[2] = 0; g2[3] = 0;
    i32x4 g3; g3[0] = 0; g3[1] = 0; g3[2] = 0; g3[3] = 0;
#if defined(__clang_major__) && (__clang_major__ >= 23)
    i32x8 g1b; for (int q = 0; q < 8; ++q) g1b[q] = 0;
    __builtin_amdgcn_tensor_load_to_lds(g0, g1, g2, g3, g1b, 0);
#else
    __builtin_amdgcn_tensor_load_to_lds(g0, g1, g2, g3, 0);
#endif
#if __has_builtin(__builtin_amdgcn_s_wait_tensorcnt)
    __builtin_amdgcn_s_wait_tensorcnt(0);
#else
    asm volatile("s_wait_tensorcnt 0x0" ::: "memory");
#endif
  }
  __syncthreads();
  dst[threadIdx.x] = buf[threadIdx.x & 63];
#else
  dst[threadIdx.x] = src[threadIdx.x];
#endif
}

// ---------------- helpers ----------------
__device__ __forceinline__ float clip01(float v) {
  return fminf(fmaxf(v, 0.0f), 1.0f);
}
__device__ __forceinline__ uint32_t hash32(uint32_t x) {
  x ^= x >> 16; x *= 0x7feb352du;
  x ^= x >> 15; x *= 0x846ca68bu;
  x ^= x >> 16;
  return x;
}
__device__ __forceinline__ float u01(uint32_t h) {
  return ((float)(h >> 8) + 1.0f) * (1.0f / 16777216.0f);  // (0,1]
}
__device__ __forceinline__ void noise3(uint32_t pix, uint32_t kx, uint32_t ky,
                                       float n[3]) {
  uint32_t h1 = hash32(pix ^ kx);
  uint32_t h2 = hash32(pix ^ ky ^ 0x9E3779B9u);
  uint32_t h3 = hash32((pix + 0x85EBCA6Bu) ^ kx);
  uint32_t h4 = hash32((pix * 0x9E3779B9u) ^ ky);
  float r1 = sqrtf(-2.0f * logf(u01(h1)));
  float sn, cn;
  sincosf(6.28318530718f * (float)(h2 >> 8) * (1.0f / 16777216.0f), &sn, &cn);
  n[0] = r1 * cn; n[1] = r1 * sn;
  float r2 = sqrtf(-2.0f * logf(u01(h3)));
  sincosf(6.28318530718f * (float)(h4 >> 8) * (1.0f / 16777216.0f), &sn, &cn);
  n[2] = r2 * cn;
}

// ============================================================================
// pass 3: contrast + saturation + sharpen + noise
// 32x32 tile (+1 halo) staged in LDS via one TDM 3-D tensor load per block
// ============================================================================
#define TS   32
#define THL  34            // tile + halo
__global__ __launch_bounds__(256) void k_color(const float* __restrict__ wsimg,
                                               const Params* __restrict__ pp,
                                               float* __restrict__ outimg) {
  __shared__ float tile[3 * THL * THL];   // 13872 B of the 320KB/WGP LDS

  const int b = blockIdx.z;
  const int x0 = blockIdx.x * TS, y0 = blockIdx.y * TS;
  const int tx = threadIdx.x, ty = threadIdx.y;
  const int ftid = ty * 16 + tx;

#if HAVE_TDM
  if (ftid == 0) {
    // 3-D tile: dim0 = x (stride 1), dim1 = y (stride IW), dim2 = c (stride IH*IW)
    const float* gsrc = wsimg + (size_t)b * 3 * IH * IW +
                        (size_t)(y0 - 1) * IW + (x0 - 1);
    uint64_t ga  = (uint64_t)(uintptr_t)gsrc;
    uint32_t lds = (uint32_t)(uintptr_t)(void*)&tile[0];
    u32x4 g0;
    g0[0] = 1u;                                // count=1 (valid user D#)
    g0[1] = lds;                               // lds_addr
    g0[2] = (uint32_t)ga;                      // global_addr[31:0]
    g0[3] = (uint32_t)((ga >> 32) & 0x1FFFFFFu) | (2u << 30);  // addr[56:32] | type=2
    i32x8 g1;
    g1[0] = 0x20000;                           // data_size = 4B
    g1[1] = (int)0xFFFF0000u;                  // tensor_dim0 = 0x7FFFFFFF (lo16)
    g1[2] = (int)0xFFFF7FFFu;                  // dim0 hi | tensor_dim1 lo (0x7FFFFFFF)
    g1[3] = (int)(((uint32_t)THL << 16) | 0x7FFFu);   // dim1 hi | tile_dim0 = 34
    g1[4] = (int)((3u << 16) | (uint32_t)THL);        // tile_dim1 = 34 | tile_dim2 = 3
    g1[5] = IW;                                // tensor_dim0_stride lo32 = 512
    g1[6] = 0;                                 // stride0 hi16 | stride1 lo16 (0x40000 lo16 = 0)
    g1[7] = (int)((uint32_t)(IH * IW) >> 16);  // stride1[47:16] = 4
    i32x4 g2; g2[0] = 0x7FFFFFFF; g2[1] = 0; g2[2] = 0; g2[3] = 0;
    i32x4 g3; g3[0] = 0; g3[1] = 0; g3[2] = 0; g3[3] = 0;
#if defined(__clang_major__) && (__clang_major__ >= 23)
    i32x8 g1b; for (int q = 0; q < 8; ++q) g1b[q] = 0;
    __builtin_amdgcn_tensor_load_to_lds(g0, g1, g2, g3, g1b, 0);
#else
    __builtin_amdgcn_tensor_load_to_lds(g0, g1, g2, g3, 0);
#endif
#if __has_builtin(__builtin_amdgcn_s_wait_tensorcnt)
    __builtin_amdgcn_s_wait_tensorcnt(0);
#else
    asm volatile("s_wait_tensorcnt 0x0" ::: "memory");
#endif
  }
#else
  // cooperative fallback: 3468 elements, clamped halo (border halo is unused)
  for (int i = ftid; i < 3 * THL * THL; i += 256) {
    int c = i / (THL * THL);
    int rem = i % (THL * THL);
    int yy = rem / THL, xx = rem % THL;
    int gy = min(max(y0 - 1 + yy, 0), IH - 1);
    int gx = min(max(x0 - 1 + xx, 0), IW - 1);
    tile[i] = wsimg[((size_t)(b * 3 + c) * IH + gy) * IW + gx];
  }
#endif
  __syncthreads();

  const bool gc  = (pp->gateCon >> b) & 1u;
  const bool gs  = (pp->gateSat >> b) & 1u;
  const bool gsh = (pp->gateSha >> b) & 1u;
  const bool gn  = (pp->gateNoi >> b) & 1u;
  const float mean = pp->means[b];
  const float cc = pp->cscale, ssat = pp->sscale, shs = pp->shscale;

  // pointwise contrast+saturation chain at a halo-local position
  auto pcs = [&](int yy, int xx, float o[3]) {
    float v0 = tile[(0 * THL + yy) * THL + xx];
    float v1 = tile[(1 * THL + yy) * THL + xx];
    float v2 = tile[(2 * THL + yy) * THL + xx];
    if (gc) {
      v0 = clip01(cc * v0 + (1.0f - cc) * mean);
      v1 = clip01(cc * v1 + (1.0f - cc) * mean);
      v2 = clip01(cc * v2 + (1.0f - cc) * mean);
    }
    if (gs) {
      float g = 0.2989f * v0 + 0.587f * v1 + 0.114f * v2;
      v0 = clip01(ssat * v0 + (1.0f - ssat) * g);
      v1 = clip01(ssat * v1 + (1.0f - ssat) * g);
      v2 = clip01(ssat * v2 + (1.0f - ssat) * g);
    }
    o[0] = v0; o[1] = v1; o[2] = v2;
  };

  // each thread produces a 2x2 quad: quadrant layout keeps stores coalesced
#pragma unroll
  for (int py = 0; py < 2; ++py) {
#pragma unroll
    for (int px = 0; px < 2; ++px) {
      const int lx = tx + 16 * px, ly = ty + 16 * py;   // 0..31 tile-local
      const int hx = lx + 1, hy = ly + 1;               // halo coords
      const int gx = x0 + lx, gy = y0 + ly;

      float ctr[3];
      pcs(hy, hx, ctr);
      float res[3] = {ctr[0], ctr[1], ctr[2]};

      if (gsh) {
        bool interior = (gx >= 1) & (gx < IW - 1) & (gy >= 1) & (gy < IH - 1);
        if (interior) {
          float acc[3] = {0.f, 0.f, 0.f};
#pragma unroll
          for (int dy = -1; dy <= 1; ++dy)
#pragma unroll
            for (int dx = -1; dx <= 1; ++dx) {
              float v[3];
              pcs(hy + dy, hx + dx, v);
              float w = (dx == 0 && dy == 0) ? 5.0f : 1.0f;
              acc[0] += w * v[0]; acc[1] += w * v[1]; acc[2] += w * v[2];
            }
#pragma unroll
          for (int ch = 0; ch < 3; ++ch) {
            float bl = clip01(acc[ch] * (1.0f / 13.0f));
            res[ch] = clip01(shs * ctr[ch] + (1.0f - shs) * bl);
          }
        }
        // border: blend(img, img, sh) == img (already clipped)
      }

      if (gn) {
        uint32_t pix = ((uint32_t)(b * IH + gy)) * IW + (uint32_t)gx;
        float n[3];
        noise3(pix, pp->nkx, pp->nky, n);
#pragma unroll
        for (int ch = 0; ch < 3; ++ch) res[ch] += n[ch] * NOISE_STD;  // no clip
      }

#pragma unroll
      for (int ch = 0; ch < 3; ++ch)
        __builtin_nontemporal_store(
            res[ch] * 2.0f - 1.0f,
            &outimg[((size_t)(b * 3 + ch) * IH + gy) * IW + gx]);
    }
  }
}

// ---------------- threefry2x32 (exact, 20 rounds) ----------------
__device__ __forceinline__ void tf2(uint32_t k0, uint32_t k1,
                                    uint32_t c0, uint32_t c1,
                                    uint32_t& o0, uint32_t& o1) {
  uint32_t ks2 = k0 ^ k1 ^ 0x1BD11BDAu;
  uint32_t x0 = c0 + k0, x1 = c1 + k1;
#define TFR(r) { x0 += x1; x1 = (x1 << (r)) | (x1 >> (32 - (r))); x1 ^= x0; }
  TFR(13) TFR(15) TFR(26) TFR(6)
  x0 += k1;  x1 += ks2 + 1u;
  TFR(17) TFR(29) TFR(16) TFR(24)
  x0 += ks2; x1 += k0 + 2u;
  TFR(13) TFR(15) TFR(26) TFR(6)
  x0 += k0;  x1 += k1 + 3u;
  TFR(17) TFR(29) TFR(16) TFR(24)
  x0 += k1;  x1 += ks2 + 4u;
  TFR(13) TFR(15) TFR(26) TFR(6)
  x0 += ks2; x1 += k0 + 5u;
#undef TFR
  o0 = x0; o1 = x1;
}

__device__ __forceinline__ float tf_uniform(uint32_t kx, uint32_t ky) {
  uint32_t o0, o1;
  tf2(kx, ky, 0u, 0u, o0, o1);
  return (float)(o0 >> 9) * (1.0f / 8388608.0f);          // [0,1)
}

__device__ __forceinline__ float tf_normal(uint32_t kx, uint32_t ky) {
  uint32_t o0, o1;
  tf2(kx, ky, 0u, 0u, o0, o1);
  float u1 = ((float)(o0 >> 9) + 0.5f) * (1.0f / 8388608.0f);  // (0,1)
  float u2 = (float)(o1 >> 9) * (1.0f / 8388608.0f);
  return sqrtf(-2.0f * logf(u1)) * cosf(6.28318530718f * u2);
}

// gate: split key -> (k1,k2); bern = 32 low bits from k1; u = uniform(k2)
__device__ __forceinline__ uint32_t tf_gate(uint32_t kx, uint32_t ky) {
  uint32_t a0, a1, b0, b1;
  tf2(kx, ky, 0u, 2u, a0, a1);   // pair (0,2)
  tf2(kx, ky, 1u, 3u, b0, b1);   // pair (1,3)
  uint32_t k1x = a0, k1y = b0, k2x = a1, k2y = b1;
  float u = tf_uniform(k2x, k2y);
  bool pg = (P_GATE > u);
  uint32_t m = 0;
  if (pg) {
    for (int i = 0; i < 16; ++i) {
      uint32_t q0, q1;
      tf2(k1x, k1y, (uint32_t)i, (uint32_t)(i + 16), q0, q1);
      if (q0 & 1u) m |= (1u << i);
      if (q1 & 1u) m |= (1u << (i + 16));
    }
  }
  return m;
}

// ---------------- setup: PRNG + per-batch composed affine ----------------
__global__ void k_setup(Params* __restrict__ pp) {
  if (threadIdx.x != 0 || blockIdx.x != 0) return;

  // jax.random.key(42) -> (0, 42); split into 18 subkeys
  uint32_t arr[36];
  for (int i = 0; i < 18; ++i) {
    uint32_t o0, o1;
    tf2(0u, 42u, (uint32_t)i, (uint32_t)(i + 18), o0, o1);
    arr[i] = o0; arr[18 + i] = o1;
  }
  uint32_t kx[18], ky[18];
  for (int j = 0; j < 18; ++j) { kx[j] = arr[2 * j]; ky[j] = arr[2 * j + 1]; }

  uint32_t gF   = tf_gate(kx[0],  ky[0]);
  uint32_t gR90 = tf_gate(kx[1],  ky[1]);
  uint32_t gT   = tf_gate(kx[2],  ky[2]);
  float t = tf_uniform(kx[3], ky[3]) * XINT_MAX;
  uint32_t gRot = tf_gate(kx[4],  ky[4]);
  float rdeg = tf_uniform(kx[5], ky[5]) * ROT_MAX;

  pp->gateHue = tf_gate(kx[6], ky[6]);
  pp->hshift  = (tf_uniform(kx[7], ky[7]) - 0.5f) * HUE_MAX;
  pp->gateBri = tf_gate(kx[8], ky[8]);
  pp->bscale  = fabsf(1.0f + tf_normal(kx[9],  ky[9])  * BRI_STD);
  pp->gateCon = tf_gate(kx[10], ky[10]);
  pp->cscale  = fabsf(1.0f + tf_normal(kx[11], ky[11]) * CON_STD);
  pp->gateSat = tf_gate(kx[12], ky[12]);
  pp->sscale  = fabsf(1.0f + tf_normal(kx[13], ky[13]) * SAT_STD);
  pp->gateSha = tf_gate(kx[14], ky[14]);
  pp->shscale = fabsf(1.0f + tf_normal(kx[15], ky[15]) * SHARP_STD);
  pp->gateNoi = tf_gate(kx[16], ky[16]);
  pp->nkx = kx[17]; pp->nky = ky[17];

  // compose per-batch affine: M = A_flip o A_rot90 o A_trans o A_rot
  const float th = rdeg * 0.017453292519943295f;
  const float cs = cosf(th), sn = sinf(th);
  const float cx = (IW - 1) * 0.5f, cy = (IH - 1) * 0.5f;
  for (int b = 0; b < NB; ++b) {
    float a = 1, bb = 0, c = 0, d = 0, e = 1, f = 0;
    if ((gRot >> b) & 1u) {
      a = cs; bb = sn; c = cx - cs * cx - sn * cy;
      d = -sn; e = cs; f = cy + sn * cx - cs * cy;
    }
    if ((gT >> b) & 1u) { c -= t; f -= t; }   // sx=x-t, sy=y-t composed
    if ((gR90 >> b) & 1u) {                    // out(y,x) <- in(x, W-1-y)
      float na = -d, nb = -e, ncv = (IW - 1) - f;
      float nd =  a, ne =  bb, nf = c;
      a = na; bb = nb; c = ncv; d = nd; e = ne; f = nf;
    }
    if ((gF >> b) & 1u) {                      // out(y,x) <- in(y, W-1-x)
      a = -a; bb = -bb; c = (IW - 1) - c;
    }
    pp->M[b][0] = a; pp->M[b][1] = bb; pp->M[b][2] = c;
    pp->M[b][3] = d; pp->M[b][4] = e;  pp->M[b][5] = f;
  }
}

__device__ __forceinline__ float tapv(const float* __restrict__ pl, int y, int x) {
  bool v = (x >= 0) & (x < IW) & (y >= 0) & (y < IH);
  int yc = min(max(y, 0), IH - 1), xc = min(max(x, 0), IW - 1);
  float r = pl[(size_t)yc * IW + xc];
  return v ? (r + 1.0f) * 0.5f : 0.0f;       // to [0,1] BEFORE valid-zeroing
}

// ---------------- pass 1: warp + hue + brightness + gray partials + mask ---
__global__ __launch_bounds__(256) void k_warp(const float* __restrict__ img,
                                              const float* __restrict__ msk,
                                              const Params* __restrict__ pp,
                                              float* __restrict__ wsimg,
                                              float* __restrict__ partials,
                                              float* __restrict__ outmsk) {
  const int b = blockIdx.z;
  const int x = blockIdx.x * 16 + threadIdx.x;
  const int y = blockIdx.y * 16 + threadIdx.y;

  const float m0 = pp->M[b][0], m1 = pp->M[b][1], m2 = pp->M[b][2];
  const float m3 = pp->M[b][3], m4 = pp->M[b][4], m5 = pp->M[b][5];
  const float sx = m0 * (float)x + m1 * (float)y + m2;
  const float sy = m3 * (float)x + m4 * (float)y + m5;

  // --- warp of 3 channels ---
  const float x0f = floorf(sx), y0f = floorf(sy);
  const float wx = sx - x0f, wy = sy - y0f;
  const int x0 = (int)x0f, y0 = (int)y0f;
  float c3[3];
  if (wx == 0.0f && wy == 0.0f) {
    // pure permutation (flip/rot90-only): wave-uniform fast path, 3 taps
#pragma unroll
    for (int ch = 0; ch < 3; ++ch)
      c3[ch] = tapv(img + (size_t)(b * 3 + ch) * IH * IW, y0, x0);
  } else {
#pragma unroll
    for (int ch = 0; ch < 3; ++ch) {
      const float* pl = img + (size_t)(b * 3 + ch) * IH * IW;
      float v00 = tapv(pl, y0, x0),     v01 = tapv(pl, y0, x0 + 1);
      float v10 = tapv(pl, y0 + 1, x0), v11 = tapv(pl, y0 + 1, x0 + 1);
      c3[ch] = (v00 * (1.f - wx) + v01 * wx) * (1.f - wy) +
               (v10 * (1.f - wx) + v11 * wx) * wy;
    }
  }

  // --- hue (gated) ---
  if ((pp->gateHue >> b) & 1u) {
    float r  = clip01(c3[0]), g = clip01(c3[1]), bl = clip01(c3[2]);
    float maxc = fmaxf(r, fmaxf(g, bl));
    float minc = fminf(r, fminf(g, bl));
    float cr = maxc - minc;
    float s  = cr / ((maxc == 0.0f) ? 1.0f : maxc);
    float crd = (cr == 0.0f) ? 1.0f : cr;
    float rc = (maxc - r) / crd, gc = (maxc - g) / crd, bc = (maxc - bl) / crd;
    float h = (maxc == r) ? (bc - gc)
                          : ((maxc == g) ? (2.0f + rc - bc) : (4.0f + gc - rc));
    h = (cr == 0.0f) ? 0.0f : h * (1.0f / 6.0f);
    h -= floorf(h);                        // mod 1
    h += pp->hshift; h -= floorf(h);
    // hsv -> rgb
    float v = maxc;
    float i6 = floorf(h * 6.0f);
    float fr = h * 6.0f - i6;
    float p = v * (1.0f - s), q = v * (1.0f - fr * s), tt = v * (1.0f - (1.0f - fr) * s);
    int ii = ((int)i6) % 6;
    float nr, ng, nb2;
    switch (ii) {
      case 0:  nr = v;  ng = tt; nb2 = p;  break;
      case 1:  nr = q;  ng = v;  nb2 = p;  break;
      case 2:  nr = p;  ng = v;  nb2 = tt; break;
      case 3:  nr = p;  ng = q;  nb2 = v;  break;
      case 4:  nr = tt; ng = p;  nb2 = v;  break;
      default: nr = v;  ng = p;  nb2 = q;  break;
    }
    c3[0] = nr; c3[1] = ng; c3[2] = nb2;
  }

  // --- brightness (gated) ---
  if ((pp->gateBri >> b) & 1u) {
    float bs = pp->bscale;
#pragma unroll
    for (int ch = 0; ch < 3; ++ch) c3[ch] = clip01(c3[ch] * bs);
  }

  // store intermediate (regular-temporal: re-read by pass 3, want it in L2)
#pragma unroll
  for (int ch = 0; ch < 3; ++ch)
    wsimg[((size_t)(b * 3 + ch) * IH + y) * IW + x] = c3[ch];

  // --- deterministic gray partial sum (for contrast mean) ---
  float gray = 0.2989f * c3[0] + 0.587f * c3[1] + 0.114f * c3[2];
  __shared__ float red[256];
  int tid = threadIdx.y * 16 + threadIdx.x;
  red[tid] = gray;
  __syncthreads();
#pragma unroll
  for (int s2 = 128; s2 > 0; s2 >>= 1) {
    if (tid < s2) red[tid] += red[tid + s2];
    __syncthreads();
  }
  if (tid == 0)
    partials[(size_t)b * 1024 + blockIdx.y * 32 + blockIdx.x] = red[0];

  // --- mask: nearest warp, straight to output (never re-read -> NT store) ---
  float mi = rintf(sx), mj = rintf(sy);
  bool valid = (mi >= 0.f) & (mi < (float)IW) & (mj >= 0.f) & (mj < (float)IH);
  int mx = min(max((int)mi, 0), IW - 1);
  int my = min(max((int)mj, 0), IH - 1);
  __builtin_nontemporal_store(
      valid ? msk[((size_t)b * IH + my) * IW + mx] : 0.0f,
      &outmsk[((size_t)b * IH + y) * IW + x]);
}

// ---------------- pass 2: deterministic mean reduction ----------------
__global__ __launch_bounds__(256) void k_mean(const float* __restrict__ partials,
                                              Params* __restrict__ pp) {
  const int b = blockIdx.x;
  const int t = threadIdx.x;
  float s = 0.0f;
  for (int i = t; i < 1024; i += 256) s += partials[(size_t)b * 1024 + i];
  __shared__ float red[256];
  red[t] = s;
  __syncthreads();
#pragma unroll
  for (int s2 = 128; s2 > 0; s2 >>= 1) {
    if (t < s2) red[t] += red[t + s2];
    __syncthreads();
  }
  if (t == 0) pp->means[b] = red[0] * (1.0f / ((float)IH * (float)IW));
}

// ---------------- host entry ----------------
extern "C" void kernel_launch(void* const* d_in, const int* in_sizes, int n_in,
                              void* d_out, int out_size, void* d_ws, size_t ws_size,
                              hipStream_t stream) {
  (void)in_sizes; (void)n_in; (void)out_size; (void)ws_size;
  const float* img = (const float*)d_in[0];
  const float* msk = (const float*)d_in[1];
  float* out = (float*)d_out;

  uint8_t* ws = (uint8_t*)d_ws;
  Params* pp       = (Params*)ws;
  float*  partials = (float*)(ws + WS_PARTIAL_OFF);
  float*  wsimg    = (float*)(ws + WS_IMG_OFF);

  float* outimg = out;
  float* outmsk = out + (size_t)NB * NC * IH * IW;

  k_setup<<<1, 1, 0, stream>>>(pp);

  dim3 blk(16, 16);
  dim3 grdW(IW / 16, IH / 16, NB);
  k_warp<<<grdW, blk, 0, stream>>>(img, msk, pp, wsimg, partials, outmsk);
  k_mean<<<NB, 256, 0, stream>>>(partials, pp);
  dim3 grdC(IW / TS, IH / TS, NB);
  k_color<<<grdC, blk, 0, stream>>>(wsimg, pp, outimg);
}